// SparseMax_83751862272312
// MI455X (gfx1250) — compile-verified
//
#include <hip/hip_runtime.h>
#include <cstdint>

#define K_DIM   32000
#define QVECS   (K_DIM / 4)       // 8000 float4 per row
#define THREADS 1024
#define CHUNKS  8                 // ceil(8000 / 1024)
#define NWAVES  (THREADS / 32)    // 32 waves (wave32)

typedef __attribute__((ext_vector_type(4))) float v4f;  // native vector: OK for nontemporal builtins

__global__ __launch_bounds__(THREADS, 1)
void sparsemax_row_kernel(const float* __restrict__ x, float* __restrict__ out) {
    __shared__ __align__(16) float srow[K_DIM];   // 128000 B staged row
    __shared__ float sA[NWAVES];
    __shared__ float sB[NWAVES];
    __shared__ float bc[2];

    const int t    = threadIdx.x;
    const int lane = t & 31;
    const int wid  = t >> 5;
    const float* rowp = x  + (size_t)blockIdx.x * K_DIM;
    float*       outp = out + (size_t)blockIdx.x * K_DIM;

    // ---- Stage the row into LDS with CDNA5 async global->LDS copies ----
    // ASYNCcnt path: no VGPR return data, deep pipelining of 8 B128/lane.
    // th:TH_LOAD_NT — streamed once, never re-read: keep it out of L2.
    #pragma unroll
    for (int c = 0; c < CHUNKS; ++c) {
        const int q = c * THREADS + t;
        if (q < QVECS) {
            const unsigned ldsOff = (unsigned)(uintptr_t)(srow + q * 4); // low 32b = LDS byte offset
            const float*   gp     = rowp + q * 4;
            asm volatile("global_load_async_to_lds_b128 %0, %1, off th:TH_LOAD_NT"
                         :: "v"(ldsOff), "v"(gp)
                         : "memory");
        }
    }
    asm volatile("s_wait_asynccnt 0" ::: "memory");  // drain this wave's async copies
    __syncthreads();                                 // whole row visible in LDS

    // ---- Pull 32 elements/thread into registers (ds_load_b128 x8) ----
    float z[32];
    #pragma unroll
    for (int c = 0; c < CHUNKS; ++c) {
        const int q = c * THREADS + t;
        if (q < QVECS) {
            const v4f f = *reinterpret_cast<const v4f*>(srow + q * 4);
            z[4*c+0] = f.x; z[4*c+1] = f.y; z[4*c+2] = f.z; z[4*c+3] = f.w;
        } else {
            z[4*c+0] = z[4*c+1] = z[4*c+2] = z[4*c+3] = -__builtin_inff();
        }
    }

    // ---- Row max (block reduction over 32 waves) ----
    float m = -__builtin_inff();
    #pragma unroll
    for (int i = 0; i < 32; ++i) m = fmaxf(m, z[i]);
    #pragma unroll
    for (int off = 16; off > 0; off >>= 1) m = fmaxf(m, __shfl_down(m, off));
    if (lane == 0) sA[wid] = m;
    __syncthreads();
    if (wid == 0) {
        float mm = sA[lane];   // NWAVES == 32 == wave width
        #pragma unroll
        for (int off = 16; off > 0; off >>= 1) mm = fmaxf(mm, __shfl_down(mm, off));
        if (lane == 0) bc[0] = mm;
    }
    __syncthreads();
    m = bc[0];
    #pragma unroll
    for (int i = 0; i < 32; ++i) z[i] -= m;   // z_max == 0 now; padded lanes stay -inf

    // ---- Newton on f(tau) = sum(max(0, z - tau)) - 1 : tau' = (S-1)/C ----
    // Convex piecewise-linear; from tau0 = -1 convergence is monotone and
    // finite (support set stabilizes). Uniform break across the block.
    float tau = -1.0f;
    #pragma unroll 1
    for (int it = 0; it < 64; ++it) {
        float s = 0.0f, cnt = 0.0f;
        #pragma unroll
        for (int i = 0; i < 32; ++i) {
            if (z[i] > tau) { s += z[i]; cnt += 1.0f; }
        }
        #pragma unroll
        for (int off = 16; off > 0; off >>= 1) {
            s   += __shfl_down(s,   off);
            cnt += __shfl_down(cnt, off);
        }
        if (lane == 0) { sA[wid] = s; sB[wid] = cnt; }
        __syncthreads();
        if (wid == 0) {
            float a = sA[lane], b = sB[lane];
            #pragma unroll
            for (int off = 16; off > 0; off >>= 1) {
                a += __shfl_down(a, off);
                b += __shfl_down(b, off);
            }
            if (lane == 0) { bc[0] = a; bc[1] = b; }
        }
        __syncthreads();
        const float S = bc[0];
        const float C = bc[1];                   // >= 1 always (z_max = 0 > tau)
        const float tauNew = (S - 1.0f) / C;
        if (tauNew == tau) break;                // support stable -> exact fixed point
        tau = tauNew;
    }

    // ---- Emit p = max(0, z - tau): nontemporal B128 stores (never re-read) ----
    #pragma unroll
    for (int c = 0; c < CHUNKS; ++c) {
        const int q = c * THREADS + t;
        if (q < QVECS) {
            v4f o;
            o.x = fmaxf(0.0f, z[4*c+0] - tau);
            o.y = fmaxf(0.0f, z[4*c+1] - tau);
            o.z = fmaxf(0.0f, z[4*c+2] - tau);
            o.w = fmaxf(0.0f, z[4*c+3] - tau);
            __builtin_nontemporal_store(o, reinterpret_cast<v4f*>(outp + q * 4));
        }
    }
}

extern "C" void kernel_launch(void* const* d_in, const int* in_sizes, int n_in,
                              void* d_out, int out_size, void* d_ws, size_t ws_size,
                              hipStream_t stream) {
    (void)n_in; (void)out_size; (void)d_ws; (void)ws_size;
    const float* x   = (const float*)d_in[0];
    float*       out = (float*)d_out;
    const int rows = in_sizes[0] / K_DIM;   // 2048
    sparsemax_row_kernel<<<rows, THREADS, 0, stream>>>(x, out);
}